// GCN_15985868276243
// MI455X (gfx1250) — compile-verified
//
#include <hip/hip_runtime.h>
#include <hip/hip_bf16.h>
#include <cstdint>

typedef __attribute__((ext_vector_type(2))) float v2f;
typedef __attribute__((ext_vector_type(8))) float v8f;

// ---------------------------------------------------------------------------
// Degree / normalization
// ---------------------------------------------------------------------------
__global__ __launch_bounds__(256) void k_init_deg(float* __restrict__ deg, int n) {
    int i = blockIdx.x * blockDim.x + threadIdx.x;
    if (i < n) deg[i] = 1.0f;   // self-loop contributes 1 to every node's degree
}

__global__ __launch_bounds__(256) void k_deg_scatter(const int* __restrict__ dst,
                                                     float* __restrict__ deg, int e) {
    int i = blockIdx.x * blockDim.x + threadIdx.x;
    if (i < e) unsafeAtomicAdd(&deg[dst[i]], 1.0f);
}

__global__ __launch_bounds__(256) void k_finalize_dinv(float* __restrict__ deg, int n) {
    int i = blockIdx.x * blockDim.x + threadIdx.x;
    if (i < n) {
        float d = deg[i];
        deg[i] = d > 0.0f ? __frsqrt_rn(d) : 0.0f;
    }
}

// ---------------------------------------------------------------------------
// GEMM1: H[N,64] = X[N,256] @ W1^T,  W1 is [64,256] (torch Linear layout)
// One block = 16 output rows, 4 waves each owning a 16-col tile.
// V_WMMA_F32_16X16X4_F32, K swept in steps of 4 (64 WMMAs / tile).
// ---------------------------------------------------------------------------
__global__ __launch_bounds__(128) void k_gemm1_wmma(const float* __restrict__ X,
                                                    const float* __restrict__ W,
                                                    float* __restrict__ H) {
    __shared__ float Xs[16][260];              // +4 pad: conflict-free b64 lane reads
    const int rowbase = blockIdx.x * 16;
    const int tid  = threadIdx.x;
    const int wave = tid >> 5;                 // 0..3 -> col tile
    const int lane = tid & 31;

    // Stage X tile: 16x256 f32 = 1024 float4, 8 per thread, coalesced.
    #pragma unroll
    for (int i = 0; i < 8; ++i) {
        int f4 = tid + i * 128;                // 0..1023
        int r  = f4 >> 6;                      // 64 float4 per row
        int c4 = f4 & 63;
        float4 v = ((const float4*)(X + (size_t)(rowbase + r) * 256))[c4];
        Xs[r][c4 * 4 + 0] = v.x; Xs[r][c4 * 4 + 1] = v.y;
        Xs[r][c4 * 4 + 2] = v.z; Xs[r][c4 * 4 + 3] = v.w;
    }
    __syncthreads();

    const int colbase = wave * 16;
    const int l15  = lane & 15;                // A: M index, B: N index
    const int koff = (lane >> 4) * 2;          // upper half-wave carries K+2,K+3
    const float* wrow = W + (size_t)(colbase + l15) * 256 + koff;

    v8f acc = {};
    for (int k0 = 0; k0 < 256; k0 += 4) {
        v2f a = *(const v2f*)&Xs[l15][k0 + koff];         // A[M=l15][K=k0+koff..+1]
        v2f b = *(const v2f*)(wrow + k0);                 // B[K][N] = W1[N][K]
        acc = __builtin_amdgcn_wmma_f32_16x16x4_f32(
                  false, a, false, b, (short)0, acc, false, false);
    }

    // D layout: vgpr i -> row i (+8 for lanes 16-31), col = lane&15
    const int m_add = (lane >> 4) * 8;
    float* out = H + (size_t)rowbase * 64 + colbase + l15;
    #pragma unroll
    for (int i = 0; i < 8; ++i) out[(size_t)(i + m_add) * 64] = acc[i];
}

// ---------------------------------------------------------------------------
// GEMM2: Z[N,48] = H1[N,64] @ W2^T (Cout=47 padded to 48; col 47 masked to 0)
// 3 waves per block.
// ---------------------------------------------------------------------------
__global__ __launch_bounds__(96) void k_gemm2_wmma(const float* __restrict__ H1,
                                                   const float* __restrict__ W2,
                                                   float* __restrict__ Z) {
    __shared__ float Hs[16][68];               // +4 pad
    const int rowbase = blockIdx.x * 16;
    const int tid  = threadIdx.x;
    const int wave = tid >> 5;                 // 0..2
    const int lane = tid & 31;

    for (int f4 = tid; f4 < 256; f4 += 96) {   // 16x64 f32 = 256 float4
        int r  = f4 >> 4;
        int c4 = f4 & 15;
        float4 v = ((const float4*)(H1 + (size_t)(rowbase + r) * 64))[c4];
        Hs[r][c4 * 4 + 0] = v.x; Hs[r][c4 * 4 + 1] = v.y;
        Hs[r][c4 * 4 + 2] = v.z; Hs[r][c4 * 4 + 3] = v.w;
    }
    __syncthreads();

    const int colbase = wave * 16;
    const int l15  = lane & 15;
    const int koff = (lane >> 4) * 2;
    const int col  = colbase + l15;
    const int wr   = col < 47 ? col : 46;      // clamp index (no OOB)
    const float msk = col < 47 ? 1.0f : 0.0f;  // zero dead column via cndmask
    const float* wrow = W2 + (size_t)wr * 64 + koff;

    v8f acc = {};
    for (int k0 = 0; k0 < 64; k0 += 4) {
        v2f a = *(const v2f*)&Hs[l15][k0 + koff];
        v2f b = *(const v2f*)(wrow + k0);
        b.x *= msk; b.y *= msk;
        acc = __builtin_amdgcn_wmma_f32_16x16x4_f32(
                  false, a, false, b, (short)0, acc, false, false);
    }

    const int m_add = (lane >> 4) * 8;
    float* out = Z + (size_t)rowbase * 48 + colbase + l15;
    #pragma unroll
    for (int i = 0; i < 8; ++i) out[(size_t)(i + m_add) * 48] = acc[i];
}

// ---------------------------------------------------------------------------
// Aggregation layer 1 (Ch = 64)
// ---------------------------------------------------------------------------
__global__ __launch_bounds__(256) void k_agg1_init(const float* __restrict__ h,
                                                   const float* __restrict__ dinv,
                                                   float* __restrict__ agg, int total) {
    int i = blockIdx.x * blockDim.x + threadIdx.x;
    if (i < total) {
        float di = dinv[i >> 6];
        agg[i] = h[i] * di * di;               // self-loop term, norm = dinv^2
    }
}

__global__ __launch_bounds__(256) void k_agg1_scatter(const int* __restrict__ src,
                                                      const int* __restrict__ dst,
                                                      const float* __restrict__ dinv,
                                                      const float* __restrict__ h,
                                                      float* __restrict__ agg, int total) {
    int i = blockIdx.x * blockDim.x + threadIdx.x;
    if (i >= total) return;
    int e = i >> 6, c = i & 63;
    int s = src[e], d = dst[e];
    float nrm = dinv[s] * dinv[d];
    unsafeAtomicAdd(&agg[(size_t)d * 64 + c], h[(size_t)s * 64 + c] * nrm);
}

__global__ __launch_bounds__(256) void k_relu_bias(float* __restrict__ agg,
                                                   const float* __restrict__ b, int total) {
    int i = blockIdx.x * blockDim.x + threadIdx.x;
    if (i < total) agg[i] = fmaxf(agg[i] + b[i & 63], 0.0f);
}

// ---------------------------------------------------------------------------
// Aggregation layer 2 (Cout = 47, z padded to stride 48; out stride 47)
// ---------------------------------------------------------------------------
__global__ __launch_bounds__(256) void k_agg2_init(const float* __restrict__ z,
                                                   const float* __restrict__ dinv,
                                                   float* __restrict__ out, int total) {
    int i = blockIdx.x * blockDim.x + threadIdx.x;
    if (i < total) {
        int n = i / 47, c = i - n * 47;
        float di = dinv[n];
        out[i] = z[(size_t)n * 48 + c] * di * di;
    }
}

__global__ __launch_bounds__(256) void k_agg2_scatter(const int* __restrict__ src,
                                                      const int* __restrict__ dst,
                                                      const float* __restrict__ dinv,
                                                      const float* __restrict__ z,
                                                      float* __restrict__ out, int total) {
    int i = blockIdx.x * blockDim.x + threadIdx.x;
    if (i >= total) return;
    int e = i / 47, c = i - e * 47;
    int s = src[e], d = dst[e];
    float nrm = dinv[s] * dinv[d];
    unsafeAtomicAdd(&out[(size_t)d * 47 + c], z[(size_t)s * 48 + c] * nrm);
}

__global__ __launch_bounds__(256) void k_bias_sigmoid(float* __restrict__ out,
                                                      const float* __restrict__ b, int total) {
    int i = blockIdx.x * blockDim.x + threadIdx.x;
    if (i < total) {
        int c = i % 47;
        float v = out[i] + b[c];
        out[i] = 1.0f / (1.0f + __expf(-v));
    }
}

// ---------------------------------------------------------------------------
extern "C" void kernel_launch(void* const* d_in, const int* in_sizes, int n_in,
                              void* d_out, int out_size, void* d_ws, size_t ws_size,
                              hipStream_t stream) {
    const float* x  = (const float*)d_in[0];   // [N,256]
    const int*   ei = (const int*)  d_in[1];   // [2,E]
    const float* W1 = (const float*)d_in[2];   // [64,256]
    const float* b1 = (const float*)d_in[3];   // [64]
    const float* W2 = (const float*)d_in[4];   // [47,64]
    const float* b2 = (const float*)d_in[5];   // [47]

    const int N = in_sizes[0] / 256;           // 100000 (multiple of 16)
    const int E = in_sizes[1] / 2;             // 1600000
    const int* src = ei;
    const int* dst = ei + E;
    float* out = (float*)d_out;                // [N,47]

    // Workspace layout (floats): dinv[N] | h[N*64] | agg1/h1[N*64] | z[N*48]
    float* dinv = (float*)d_ws;
    float* h    = dinv + N;
    float* agg1 = h    + (size_t)N * 64;
    float* z    = agg1 + (size_t)N * 64;

    const int B = 256;
    // 1) degrees -> dinv
    k_init_deg     <<<(N + B - 1) / B, B, 0, stream>>>(dinv, N);
    k_deg_scatter  <<<(E + B - 1) / B, B, 0, stream>>>(dst, dinv, E);
    k_finalize_dinv<<<(N + B - 1) / B, B, 0, stream>>>(dinv, N);

    // 2) h = x @ W1^T   (WMMA f32 16x16x4)
    k_gemm1_wmma<<<N / 16, 128, 0, stream>>>(x, W1, h);

    // 3) layer-1 aggregation + bias + ReLU (in place -> h1)
    int tot1 = N * 64;
    k_agg1_init   <<<(tot1 + B - 1) / B, B, 0, stream>>>(h, dinv, agg1, tot1);
    int etot1 = E * 64;
    k_agg1_scatter<<<(etot1 + B - 1) / B, B, 0, stream>>>(src, dst, dinv, h, agg1, etot1);
    k_relu_bias   <<<(tot1 + B - 1) / B, B, 0, stream>>>(agg1, b1, tot1);

    // 4) z = h1 @ W2^T  (WMMA f32 16x16x4, 48-col padded)
    k_gemm2_wmma<<<N / 16, 96, 0, stream>>>(agg1, W2, z);

    // 5) layer-2 aggregation directly into d_out, then bias + sigmoid
    int tot2 = N * 47;
    k_agg2_init   <<<(tot2 + B - 1) / B, B, 0, stream>>>(z, dinv, out, tot2);
    int etot2 = E * 47;
    k_agg2_scatter<<<(etot2 + B - 1) / B, B, 0, stream>>>(src, dst, dinv, z, out, etot2);
    k_bias_sigmoid<<<(tot2 + B - 1) / B, B, 0, stream>>>(out, b2, tot2);
}